// ParameterWrapper_9388798509068
// MI455X (gfx1250) — compile-verified
//
#include <hip/hip_runtime.h>

// out[b, r, c] = x[b, r, c] * w[c]   (pure bandwidth-bound elementwise scale)
//
// MI455X strategy:
//  - 128-bit vector memory ops end to end (global_load_b128 / global_store_b128).
//  - TH_NT on the 256MB x stream + 256MB out stream (each byte touched once);
//    weight (16KB) loaded with default temporal policy -> stays hot in WGP$/L2
//    for all 16384 rows.
//  - Compile-time trip count (N=4096 -> 1024 float4/row -> 4 iters @ 256 thr)
//    so the loop fully unrolls: no remainder path, no loop-carried SALU.

typedef __attribute__((ext_vector_type(4))) float v4f;

template <int VEC_PER_ROW>
__global__ void __launch_bounds__(256) row_scale_fixed(
    const float* __restrict__ x,
    const float* __restrict__ w,
    float* __restrict__ out)
{
    const long long row = blockIdx.x;
    const int t = threadIdx.x;

    const v4f* __restrict__ xv = (const v4f*)(x) + row * (long long)VEC_PER_ROW;
    v4f* __restrict__ ov       = (v4f*)(out)     + row * (long long)VEC_PER_ROW;
    const v4f* __restrict__ wv = (const v4f*)(w);

    static_assert(VEC_PER_ROW % 256 == 0, "row must tile evenly over 256 threads");
    #pragma unroll
    for (int k = 0; k < VEC_PER_ROW / 256; ++k) {
        const int idx = t + k * 256;
        v4f xa = __builtin_nontemporal_load(&xv[idx]);  // read-once stream
        v4f wa = wv[idx];                               // temporal: L2-resident
        v4f r  = xa * wa;
        __builtin_nontemporal_store(r, &ov[idx]);       // write-once stream
    }
}

// Generic fallback for row lengths other than 4096 floats.
__global__ void __launch_bounds__(256) row_scale_generic(
    const float* __restrict__ x,
    const float* __restrict__ w,
    float* __restrict__ out,
    int n)   // row length in floats
{
    const long long row = blockIdx.x;
    const long long base = row * (long long)n;
    for (int c = threadIdx.x; c < n; c += 256) {
        float xa = __builtin_nontemporal_load(&x[base + c]);
        float r  = xa * w[c];
        __builtin_nontemporal_store(r, &out[base + c]);
    }
}

extern "C" void kernel_launch(void* const* d_in, const int* in_sizes, int n_in,
                              void* d_out, int out_size, void* d_ws, size_t ws_size,
                              hipStream_t stream) {
    const float* x = (const float*)d_in[0];   // (4, 4096, N) fp32
    const float* w = (const float*)d_in[1];   // (N,) fp32
    float* out = (float*)d_out;               // (4, 4096, N) fp32

    const int N = in_sizes[1];                // 4096
    const long long total = (long long)in_sizes[0];
    const int rows = (int)(total / N);        // 16384

    if (N == 4096) {
        row_scale_fixed<1024><<<dim3(rows), dim3(256), 0, stream>>>(x, w, out);
    } else if (N % 1024 == 0) {
        // Count of 4096-float chunks; unused, the launch below works per full row.
        const int vrows = (int)(total / 4096);
        // Weight indexing differs per chunk only if N != 4096, so fall back.
        row_scale_generic<<<dim3(rows), dim3(256), 0, stream>>>(x, w, out, N);
        (void)vrows;
    } else {
        row_scale_generic<<<dim3(rows), dim3(256), 0, stream>>>(x, w, out, N);
    }
}